// AdvancedMolecularGNN_74938589381413
// MI455X (gfx1250) — compile-verified
//
#include <hip/hip_runtime.h>

#define HIDF 128
#define HEADS 4
#define NEG_SLOPE 0.2f
#define BN_EPS 1e-5f

typedef __attribute__((ext_vector_type(2))) float v2f;
typedef __attribute__((ext_vector_type(8))) float v8f;

// ---------------------------------------------------------------------------
// GEMM: C[M x Kout] = A[M x 128] * W[128 x Kout], fp32 WMMA 16x16x4.
// Block = 256 threads = 8 waves; wave w owns column tile (blockIdx.y*128+w*16).
// B fragments (32 steps x v2f = 64 VGPRs) preloaded once per wave, then the
// block grid-strides over 16-row tiles of A, so W traffic is fully amortized.
// Fragment layout per CDNA5 ISA 7.12.2:
//   A 16x4 f32 : lanes 0-15 row M=lane, (v0,v1)=(K=0,K=1); lanes16-31 (K=2,K=3)
//   B 4x16 f32 : lanes 0-15 col N=lane, (v0,v1)=(K=0,K=1); lanes16-31 (K=2,K=3)
//   C 16x16 f32: vgpr j -> row m0 + 8*(lane>>4) + j, col n0 + (lane&15)
// ---------------------------------------------------------------------------
__global__ __launch_bounds__(256) void gemm128_wmma(
    const float* __restrict__ A, const float* __restrict__ W,
    float* __restrict__ C, int M, int Kout) {
  const int lane = threadIdx.x & 31;
  const int wave = threadIdx.x >> 5;
  const int half = lane >> 4;
  const int l16  = lane & 15;
  const int n    = blockIdx.y * 128 + wave * 16 + l16;

  v2f bfrag[32];
#pragma unroll
  for (int s = 0; s < 32; ++s) {
    const int k = 4 * s + 2 * half;
    bfrag[s].x = W[(size_t)k * Kout + n];
    bfrag[s].y = W[(size_t)(k + 1) * Kout + n];
  }

  const int mtiles = M >> 4;
  for (int mt = blockIdx.x; mt < mtiles; mt += gridDim.x) {
    const int m0 = mt << 4;
    const float* arow = A + (size_t)(m0 + l16) * HIDF + 2 * half;
    v8f acc = {};
#pragma unroll
    for (int s = 0; s < 32; ++s) {
      const v2f a = *(const v2f*)(arow + 4 * s);
      acc = __builtin_amdgcn_wmma_f32_16x16x4_f32(
          /*neg_a=*/false, a, /*neg_b=*/false, bfrag[s],
          /*c_mod=*/(short)0, acc, /*reuse_a=*/false, /*reuse_b=*/false);
    }
    float* crow = C + (size_t)(m0 + half * 8) * Kout + n;
#pragma unroll
    for (int j = 0; j < 8; ++j) crow[(size_t)j * Kout] = acc[j];
  }
}

// ---------------------------------------------------------------------------
// Small helpers
// ---------------------------------------------------------------------------
__global__ void fill_f32(float* __restrict__ p, float v, long long n) {
  const long long i = (long long)blockIdx.x * blockDim.x + threadIdx.x;
  if (i < n) p[i] = v;
}

__global__ void deg_edges(const int* __restrict__ ei, int E, float* __restrict__ deg) {
  const int e = blockIdx.x * blockDim.x + threadIdx.x;
  if (e < E) atomicAdd(&deg[ei[E + e]], 1.0f);
}

__global__ void rsqrt_ip(float* __restrict__ p, int n) {
  const int i = blockIdx.x * blockDim.x + threadIdx.x;
  if (i < n) p[i] = rsqrtf(p[i]);
}

__device__ inline void atomicMaxF(float* addr, float val) {
  unsigned int* a = (unsigned int*)addr;
  unsigned int cur = __float_as_uint(*addr);
  while (__uint_as_float(cur) < val) {
    const unsigned int prev = atomicCAS(a, cur, __float_as_uint(val));
    if (prev == cur) break;
    cur = prev;
  }
}

// ---------------------------------------------------------------------------
// GCN: agg[dst] += t[src] * dinv[src]*dinv[dst].  One wave per edge (incl.
// self loops e>=E), float4 gathers, 4 f32 atomics per lane (128 feat / wave).
// ---------------------------------------------------------------------------
__global__ void gcn_scatter(const float* __restrict__ t, const float* __restrict__ dinv,
                            const int* __restrict__ ei, int E, int N,
                            float* __restrict__ agg) {
  const int lane = threadIdx.x & 31;
  const long long w = ((long long)blockIdx.x * blockDim.x + threadIdx.x) >> 5;
  if (w >= (long long)E + N) return;
  int s, d;
  if (w < E) { s = ei[w]; d = ei[E + w]; } else { s = d = (int)(w - E); }
  const float en = dinv[s] * dinv[d];
  const float4 v = *(const float4*)(t + (size_t)s * HIDF + lane * 4);
  float* o = agg + (size_t)d * HIDF + lane * 4;
  atomicAdd(o + 0, v.x * en);
  atomicAdd(o + 1, v.y * en);
  atomicAdd(o + 2, v.z * en);
  atomicAdd(o + 3, v.w * en);
}

__global__ void bn_relu(const float* __restrict__ agg, const float* __restrict__ b,
                        const float* __restrict__ gamma, const float* __restrict__ beta,
                        const float* __restrict__ mean, const float* __restrict__ var,
                        float* __restrict__ h, int residual, int N) {
  const long long idx = (long long)blockIdx.x * blockDim.x + threadIdx.x;
  if (idx >= (long long)N * HIDF) return;
  const int c = (int)(idx & (HIDF - 1));
  float v = agg[idx] + b[c];
  v = (v - mean[c]) * rsqrtf(var[c] + BN_EPS) * gamma[c] + beta[c];
  v = fmaxf(v, 0.0f);
  h[idx] = residual ? (h[idx] + v) : v;
}

// ---------------------------------------------------------------------------
// GAT: per-node attention scores a_s/a_d (wave per node, shfl reduction).
// ---------------------------------------------------------------------------
__global__ void gat_scores(const float* __restrict__ hh,
                           const float* __restrict__ atts, const float* __restrict__ attd,
                           float* __restrict__ a_s, float* __restrict__ a_d, int N) {
  const int lane = threadIdx.x & 31;
  const long long n = ((long long)blockIdx.x * blockDim.x + threadIdx.x) >> 5;
  if (n >= N) return;
#pragma unroll
  for (int hd = 0; hd < HEADS; ++hd) {
    const float4 v  = *(const float4*)(hh + (size_t)n * (HEADS * HIDF) + hd * HIDF + lane * 4);
    const float4 ws = *(const float4*)(atts + hd * HIDF + lane * 4);
    const float4 wd = *(const float4*)(attd + hd * HIDF + lane * 4);
    float ss = v.x * ws.x + v.y * ws.y + v.z * ws.z + v.w * ws.w;
    float sd = v.x * wd.x + v.y * wd.y + v.z * wd.z + v.w * wd.w;
#pragma unroll
    for (int off = 16; off > 0; off >>= 1) {
      ss += __shfl_xor(ss, off, 32);
      sd += __shfl_xor(sd, off, 32);
    }
    if (lane == 0) { a_s[n * HEADS + hd] = ss; a_d[n * HEADS + hd] = sd; }
  }
}

__global__ void gat_logits_max(const float* __restrict__ a_s, const float* __restrict__ a_d,
                               const int* __restrict__ ei, int E, int N,
                               float* __restrict__ ebuf, float* __restrict__ emax) {
  const long long idx = (long long)blockIdx.x * blockDim.x + threadIdx.x;
  if (idx >= ((long long)E + N) * HEADS) return;
  const int e = (int)(idx >> 2), hd = (int)(idx & 3);
  int s, d;
  if (e < E) { s = ei[e]; d = ei[E + e]; } else { s = d = e - E; }
  float v = a_s[s * HEADS + hd] + a_d[d * HEADS + hd];
  v = v > 0.0f ? v : NEG_SLOPE * v;
  ebuf[idx] = v;
  atomicMaxF(&emax[d * HEADS + hd], v);
}

__global__ void gat_exp_sum(const int* __restrict__ ei, int E, int N,
                            const float* __restrict__ emax,
                            float* __restrict__ ebuf, float* __restrict__ denom) {
  const long long idx = (long long)blockIdx.x * blockDim.x + threadIdx.x;
  if (idx >= ((long long)E + N) * HEADS) return;
  const int e = (int)(idx >> 2), hd = (int)(idx & 3);
  const int d = (e < E) ? ei[E + e] : (e - E);
  const float ex = __expf(ebuf[idx] - emax[d * HEADS + hd]);
  ebuf[idx] = ex;
  atomicAdd(&denom[d * HEADS + hd], ex);
}

__global__ void newh_init(const float* __restrict__ h, const float* __restrict__ gb,
                          float* __restrict__ nh, int N) {
  const long long idx = (long long)blockIdx.x * blockDim.x + threadIdx.x;
  if (idx >= (long long)N * HIDF) return;
  nh[idx] = h[idx] + gb[idx & (HIDF - 1)];
}

// newh[dst] += 0.25 * alpha * hh[src, head, :]  (mean over heads folded in)
__global__ void gat_aggregate(const float* __restrict__ hh, const float* __restrict__ ebuf,
                              const float* __restrict__ denom, const int* __restrict__ ei,
                              int E, int N, float* __restrict__ nh) {
  const int lane = threadIdx.x & 31;
  const long long w = ((long long)blockIdx.x * blockDim.x + threadIdx.x) >> 5;
  if (w >= ((long long)E + N) * HEADS) return;
  const int e = (int)(w >> 2), hd = (int)(w & 3);
  int s, d;
  if (e < E) { s = ei[e]; d = ei[E + e]; } else { s = d = e - E; }
  const float alpha = ebuf[w] * 0.25f / (denom[d * HEADS + hd] + 1e-16f);
  const float4 v = *(const float4*)(hh + (size_t)s * (HEADS * HIDF) + hd * HIDF + lane * 4);
  float* o = nh + (size_t)d * HIDF + lane * 4;
  atomicAdd(o + 0, v.x * alpha);
  atomicAdd(o + 1, v.y * alpha);
  atomicAdd(o + 2, v.z * alpha);
  atomicAdd(o + 3, v.w * alpha);
}

// ---------------------------------------------------------------------------
// Global mean pool + classifier
// ---------------------------------------------------------------------------
__global__ void pool_scatter(const float* __restrict__ nh, const int* __restrict__ batch,
                             float* __restrict__ gsum, float* __restrict__ cnt, int N) {
  const long long idx = (long long)blockIdx.x * blockDim.x + threadIdx.x;
  if (idx >= (long long)N * HIDF) return;
  const int n = (int)(idx >> 7), c = (int)(idx & (HIDF - 1));
  const int g = batch[n];
  atomicAdd(&gsum[(size_t)g * HIDF + c], nh[idx]);
  if (c == 0) atomicAdd(&cnt[g], 1.0f);
}

__global__ void mlp1(const float* __restrict__ gsum, const float* __restrict__ cnt,
                     const float* __restrict__ W, const float* __restrict__ b,
                     float* __restrict__ z, int G) {
  const int idx = blockIdx.x * blockDim.x + threadIdx.x;
  if (idx >= G * 64) return;
  const int g = idx >> 6, j = idx & 63;
  const float inv = 1.0f / fmaxf(cnt[g], 1.0f);
  float acc = b[j];
  for (int c = 0; c < HIDF; ++c) acc += gsum[(size_t)g * HIDF + c] * inv * W[c * 64 + j];
  z[idx] = fmaxf(acc, 0.0f);
}

__global__ void mlp2(const float* __restrict__ z1, const float* __restrict__ W,
                     const float* __restrict__ b, float* __restrict__ z, int G) {
  const int idx = blockIdx.x * blockDim.x + threadIdx.x;
  if (idx >= G * 32) return;
  const int g = idx >> 5, j = idx & 31;
  float acc = b[j];
  for (int c = 0; c < 64; ++c) acc += z1[g * 64 + c] * W[c * 32 + j];
  z[idx] = fmaxf(acc, 0.0f);
}

__global__ void mlp3(const float* __restrict__ z2, const float* __restrict__ W,
                     const float* __restrict__ b, float* __restrict__ out, int G) {
  const int g = blockIdx.x * blockDim.x + threadIdx.x;
  if (g >= G) return;
  float acc = b[0];
  for (int c = 0; c < 32; ++c) acc += z2[g * 32 + c] * W[c];
  out[g] = acc;
}

// ---------------------------------------------------------------------------
extern "C" void kernel_launch(void* const* d_in, const int* in_sizes, int n_in,
                              void* d_out, int out_size, void* d_ws, size_t ws_size,
                              hipStream_t stream) {
  const float* x     = (const float*)d_in[0];
  const int*   ei    = (const int*)d_in[1];
  const int*   batch = (const int*)d_in[2];
  const float* gcn_W = (const float*)d_in[3];
  const float* gcn_b = (const float*)d_in[4];
  const float* bn_g  = (const float*)d_in[5];
  const float* bn_be = (const float*)d_in[6];
  const float* bn_m  = (const float*)d_in[7];
  const float* bn_v  = (const float*)d_in[8];
  const float* gat_W = (const float*)d_in[9];
  const float* att_s = (const float*)d_in[10];
  const float* att_d = (const float*)d_in[11];
  const float* gat_b = (const float*)d_in[12];
  const float* c1W   = (const float*)d_in[13];
  const float* c1b   = (const float*)d_in[14];
  const float* c2W   = (const float*)d_in[15];
  const float* c2b   = (const float*)d_in[16];
  const float* c3W   = (const float*)d_in[17];
  const float* c3b   = (const float*)d_in[18];
  float* out = (float*)d_out;

  const int N = in_sizes[0] / HIDF;
  const int E = in_sizes[1] / 2;
  const int G = out_size;
  (void)n_in; (void)ws_size;

  // workspace layout (fp32): ~323 MB total
  float* ws = (float*)d_ws;
  size_t o = 0;
  float* f_t    = ws + o; o += (size_t)N * HEADS * HIDF;   // GEMM out / GAT hh
  float* f_h    = ws + o; o += (size_t)N * HIDF;           // node features
  float* f_agg  = ws + o; o += (size_t)N * HIDF;           // agg / final newh
  float* f_dinv = ws + o; o += (size_t)N;
  float* f_as   = ws + o; o += (size_t)N * HEADS;
  float* f_ad   = ws + o; o += (size_t)N * HEADS;
  float* f_emax = ws + o; o += (size_t)N * HEADS;
  float* f_den  = ws + o; o += (size_t)N * HEADS;
  float* f_eb   = ws + o; o += (size_t)(E + N) * HEADS;
  float* f_gsum = ws + o; o += (size_t)G * HIDF;
  float* f_cnt  = ws + o; o += (size_t)G;
  float* f_z1   = ws + o; o += (size_t)G * 64;
  float* f_z2   = ws + o; o += (size_t)G * 32;

  const dim3 blk(256);
  auto cdiv = [](long long a, long long b) { return (unsigned)((a + b - 1) / b); };
  const long long NE = (long long)E + N;

  // degree (with self loops) -> dinv = rsqrt(deg)
  fill_f32<<<cdiv(N, 256), blk, 0, stream>>>(f_dinv, 1.0f, N);
  deg_edges<<<cdiv(E, 256), blk, 0, stream>>>(ei, E, f_dinv);
  rsqrt_ip<<<cdiv(N, 256), blk, 0, stream>>>(f_dinv, N);

  // 3x GCN layers
  for (int i = 0; i < 3; ++i) {
    const float* hin = (i == 0) ? x : f_h;
    gemm128_wmma<<<dim3(1024, 1), blk, 0, stream>>>(hin, gcn_W + (size_t)i * HIDF * HIDF,
                                                    f_t, N, HIDF);
    hipMemsetAsync(f_agg, 0, (size_t)N * HIDF * sizeof(float), stream);
    gcn_scatter<<<cdiv(NE * 32, 256), blk, 0, stream>>>(f_t, f_dinv, ei, E, N, f_agg);
    bn_relu<<<cdiv((long long)N * HIDF, 256), blk, 0, stream>>>(
        f_agg, gcn_b + i * HIDF, bn_g + i * HIDF, bn_be + i * HIDF,
        bn_m + i * HIDF, bn_v + i * HIDF, f_h, i > 0, N);
  }

  // GAT (4 heads, mean)
  gemm128_wmma<<<dim3(1024, HEADS), blk, 0, stream>>>(f_h, gat_W, f_t, N, HEADS * HIDF);
  gat_scores<<<cdiv((long long)N * 32, 256), blk, 0, stream>>>(f_t, att_s, att_d, f_as, f_ad, N);
  fill_f32<<<cdiv((long long)N * HEADS, 256), blk, 0, stream>>>(f_emax, -1e30f, (long long)N * HEADS);
  hipMemsetAsync(f_den, 0, (size_t)N * HEADS * sizeof(float), stream);
  gat_logits_max<<<cdiv(NE * HEADS, 256), blk, 0, stream>>>(f_as, f_ad, ei, E, N, f_eb, f_emax);
  gat_exp_sum<<<cdiv(NE * HEADS, 256), blk, 0, stream>>>(ei, E, N, f_emax, f_eb, f_den);
  newh_init<<<cdiv((long long)N * HIDF, 256), blk, 0, stream>>>(f_h, gat_b, f_agg, N);
  gat_aggregate<<<cdiv(NE * HEADS * 32, 256), blk, 0, stream>>>(f_t, f_eb, f_den, ei, E, N, f_agg);

  // pool + classifier
  hipMemsetAsync(f_gsum, 0, (size_t)G * HIDF * sizeof(float), stream);
  hipMemsetAsync(f_cnt, 0, (size_t)G * sizeof(float), stream);
  pool_scatter<<<cdiv((long long)N * HIDF, 256), blk, 0, stream>>>(f_agg, batch, f_gsum, f_cnt, N);
  mlp1<<<cdiv((long long)G * 64, 256), blk, 0, stream>>>(f_gsum, f_cnt, c1W, c1b, f_z1, G);
  mlp2<<<cdiv((long long)G * 32, 256), blk, 0, stream>>>(f_z1, c2W, c2b, f_z2, G);
  mlp3<<<cdiv(G, 256), blk, 0, stream>>>(f_z2, c3W, c3b, out, G);
}